// YOLOV7Loss_76287209111906
// MI455X (gfx1250) — compile-verified
//
#include <hip/hip_runtime.h>

#define EPS_ 1e-7f
#define FOUR_OVER_PI2 0.40528473456935108577f

typedef __attribute__((ext_vector_type(2))) float v2f;
typedef __attribute__((ext_vector_type(8))) float v8f;

static __device__ __forceinline__ float sigm_(float x) {
    return 1.0f / (1.0f + expf(-x));
}
// numerically-stable softplus == BCEWithLogits(x, 0)
static __device__ __forceinline__ float softplus_(float x) {
    return fmaxf(x, 0.0f) + log1pf(expf(-fabsf(x)));
}

// Wave32 sum-reduction via V_WMMA_F32_16X16X4_F32 with all-ones B matrix.
// A layout (16x4 f32): lanes 0-15 hold A[M=lane, K=0..1] in vgpr0/1,
// lanes 16-31 hold A[M=lane-16, K=2..3]. Put v in component 0, 0 in comp 1:
// rowsum(m) = v(lane m) + v(lane m+16). D layout: vgpr p on lanes 0-15 is
// M=p, on lanes 16-31 is M=p+8, so sum of 8 vgprs gives half-sum per lane
// group; one shfl_xor(16) completes the 32-lane total on every lane.
// EXEC must be all ones at the call site (callers mask contributions, never
// early-return).
static __device__ __forceinline__ float wave_sum32(float v) {
    v2f a = { v, 0.0f };
    v2f ones = { 1.0f, 1.0f };
    v8f c = {};
    v8f d = __builtin_amdgcn_wmma_f32_16x16x4_f32(
        /*neg_a=*/false, a, /*neg_b=*/false, ones,
        /*c_mod=*/(short)0, c, /*reuse_a=*/false, /*reuse_b=*/false);
    float s = d[0] + d[1] + d[2] + d[3] + d[4] + d[5] + d[6] + d[7];
    s += __shfl_xor(s, 16, 32);
    return s;
}

static __device__ __forceinline__ float ciou_(float cx1, float cy1, float w1, float h1,
                                              float cx2, float cy2, float w2, float h2) {
    float x1a = cx1 - 0.5f * w1, y1a = cy1 - 0.5f * h1;
    float x2a = cx1 + 0.5f * w1, y2a = cy1 + 0.5f * h1;
    float x1b = cx2 - 0.5f * w2, y1b = cy2 - 0.5f * h2;
    float x2b = cx2 + 0.5f * w2, y2b = cy2 + 0.5f * h2;
    float iw = fmaxf(fminf(x2a, x2b) - fmaxf(x1a, x1b), 0.0f);
    float ih = fmaxf(fminf(y2a, y2b) - fmaxf(y1a, y1b), 0.0f);
    float inter = iw * ih;
    float uni = w1 * h1 + w2 * h2 - inter;
    float iou = inter / (uni + EPS_);
    float cw = fmaxf(x2a, x2b) - fminf(x1a, x1b);
    float ch = fmaxf(y2a, y2b) - fminf(y1a, y1b);
    float diag = cw * cw + ch * ch + EPS_;
    float cent = (cx1 - cx2) * (cx1 - cx2) + (cy1 - cy2) * (cy1 - cy2);
    float at = atanf(w1 / (h1 + EPS_)) - atanf(w2 / (h2 + EPS_));
    float v = FOUR_OVER_PI2 * at * at;
    float alpha = v / (v - iou + 1.0f + EPS_);
    return iou - (cent / diag + v * alpha);
}

// ---------------------------------------------------------------------------
// Kernels
// ---------------------------------------------------------------------------

__global__ void yolo_zero_ws(double* acc, float* tobj, int ftot) {
    int i = blockIdx.x * blockDim.x + threadIdx.x;
    if (i < 9) acc[i] = 0.0;
    if (i < ftot) tobj[i] = 0.0f;
}

// Per-target work for one scale: gather box+cls logits, CIoU, scatter tobj,
// wave-reduce lbox/lcls contributions with WMMA, f64-atomic into accumulators.
__global__ void yolo_target_kernel(const float* __restrict__ box,
                                   const float* __restrict__ cls,
                                   const int* __restrict__ bi,
                                   const int* __restrict__ ai,
                                   const int* __restrict__ gyi,
                                   const int* __restrict__ gxi,
                                   const int* __restrict__ tci,
                                   const float* __restrict__ tbx,
                                   const float* __restrict__ anc,
                                   float* __restrict__ tobj,
                                   double* __restrict__ acc_lbox,
                                   double* __restrict__ acc_lcls,
                                   int n, int h, int w) {
    int idx = blockIdx.x * blockDim.x + threadIdx.x;
    bool valid = idx < n;
    int j = valid ? idx : 0;                 // clamp so masked lanes load safely

    int b  = bi[j];
    int a  = ai[j];
    int gy = gyi[j];
    int gx = gxi[j];
    int tc = tci[j];
    long base = (((long)b * 3 + a) * h + gy) * (long)w + gx;

    float4 pb = *(const float4*)(box + base * 4);        // 16B aligned
    float2 an = *(const float2*)(anc + (long)j * 2);     // 8B aligned
    float4 tb = *(const float4*)(tbx + (long)j * 4);     // 16B aligned

    float px = sigm_(pb.x) * 2.0f - 0.5f;
    float py = sigm_(pb.y) * 2.0f - 0.5f;
    float sw = sigm_(pb.z) * 2.0f;
    float sh = sigm_(pb.w) * 2.0f;
    float pw = sw * sw * an.x;
    float ph = sh * sh * an.y;

    float ciou = ciou_(px, py, pw, ph, tb.x, tb.y, tb.z, tb.w);

    if (valid) tobj[base] = fmaxf(ciou, 0.0f);           // .at[].set: some writer wins

    // class BCE vs one-hot: sum_c softplus(x_c) - x_{tcls}
    const float* cp = cls + base * 80;
    float csum = 0.0f, xt = 0.0f;
    #pragma unroll
    for (int c = 0; c < 80; c += 4) {
        float4 x = *(const float4*)(cp + c);
        csum += softplus_(x.x) + softplus_(x.y) + softplus_(x.z) + softplus_(x.w);
        if (tc >= c && tc < c + 4) {
            xt = (tc == c) ? x.x : (tc == c + 1) ? x.y : (tc == c + 2) ? x.z : x.w;
        }
    }

    float lbox_c = valid ? (1.0f - ciou) : 0.0f;
    float lcls_c = valid ? (csum - xt) : 0.0f;

    float wb = wave_sum32(lbox_c);     // EXEC all ones here (no early returns)
    float wc = wave_sum32(lcls_c);
    if ((threadIdx.x & 31) == 0) {
        atomicAdd(acc_lbox, (double)wb);
        atomicAdd(acc_lcls, (double)wc);
    }
}

// Dense objectness BCE over the whole grid: sum softplus(x) - x*tobj.
__global__ void yolo_obj_kernel(const float* __restrict__ cnf,
                                const float* __restrict__ tobj,
                                double* __restrict__ acc, long m) {
    long stride = (long)gridDim.x * blockDim.x;
    float s = 0.0f;
    for (long j = (long)blockIdx.x * blockDim.x + threadIdx.x; j < m; j += stride) {
        float x = cnf[j];
        s += softplus_(x) - x * tobj[j];
    }
    float ws = wave_sum32(s);          // all lanes reconverged after the loop
    if ((threadIdx.x & 31) == 0) atomicAdd(acc, (double)ws);
}

__global__ void yolo_finalize(const double* __restrict__ acc, float* __restrict__ out) {
    if (threadIdx.x != 0 || blockIdx.x != 0) return;
    double lbox = acc[0] / 6000.0 + acc[1] / 4000.0 + acc[2] / 2000.0;
    double lcls = acc[3] / (6000.0 * 80.0) + acc[4] / (4000.0 * 80.0) + acc[5] / (2000.0 * 80.0);
    double lobj = (acc[6] / 614400.0) * 4.0 + (acc[7] / 153600.0) * 1.0 + (acc[8] / 38400.0) * 0.4;
    out[0] = (float)(lbox * 0.05 * 32.0);
    out[1] = (float)(lobj * 0.70 * 32.0);
    out[2] = (float)(lcls * 0.30 * 32.0);
}

// ---------------------------------------------------------------------------
// Launch: all on `stream`, no allocations, deterministic, graph-capture safe.
// Input order (setup_inputs dict): per scale i: box,cnf,cls,b,a,gy,gx,tcls,tbox,anc
// ---------------------------------------------------------------------------
extern "C" void kernel_launch(void* const* d_in, const int* in_sizes, int n_in,
                              void* d_out, int out_size, void* d_ws, size_t ws_size,
                              hipStream_t stream) {
    (void)in_sizes; (void)n_in; (void)out_size; (void)ws_size;

    static const int  H[3] = {80, 40, 20};
    static const int  W[3] = {80, 40, 20};
    static const int  N[3] = {6000, 4000, 2000};
    static const long M[3] = {614400, 153600, 38400};   // 32*3*H*W

    double* acc       = (double*)d_ws;                  // 9 doubles: lbox[3], lcls[3], lobj[3]
    float*  tobj_base = (float*)((char*)d_ws + 128);
    const long toff[3] = {0, M[0], M[0] + M[1]};
    const int  ftot    = (int)(M[0] + M[1] + M[2]);     // 806400 floats ~ 3.2 MB

    yolo_zero_ws<<<(ftot + 255) / 256, 256, 0, stream>>>(acc, tobj_base, ftot);

    for (int i = 0; i < 3; ++i) {
        const float* box = (const float*)d_in[10 * i + 0];
        const float* cls = (const float*)d_in[10 * i + 2];
        const int*   bi  = (const int*)  d_in[10 * i + 3];
        const int*   ai  = (const int*)  d_in[10 * i + 4];
        const int*   gy  = (const int*)  d_in[10 * i + 5];
        const int*   gx  = (const int*)  d_in[10 * i + 6];
        const int*   tc  = (const int*)  d_in[10 * i + 7];
        const float* tbx = (const float*)d_in[10 * i + 8];
        const float* an  = (const float*)d_in[10 * i + 9];
        yolo_target_kernel<<<(N[i] + 255) / 256, 256, 0, stream>>>(
            box, cls, bi, ai, gy, gx, tc, tbx, an,
            tobj_base + toff[i], acc + i, acc + 3 + i, N[i], H[i], W[i]);
    }

    for (int i = 0; i < 3; ++i) {
        const float* cnf = (const float*)d_in[10 * i + 1];
        yolo_obj_kernel<<<1024, 256, 0, stream>>>(cnf, tobj_base + toff[i], acc + 6 + i, M[i]);
    }

    yolo_finalize<<<1, 32, 0, stream>>>(acc, (float*)d_out);
}